// GCN_43499428774060
// MI455X (gfx1250) — compile-verified
//
#include <hip/hip_runtime.h>
#include <hip/hip_bf16.h>

typedef __attribute__((ext_vector_type(2))) float v2f;
typedef __attribute__((ext_vector_type(8))) float v8f;

#define FEAT 128

// ---------------- degree / normalization ----------------

__global__ __launch_bounds__(256) void init_deg_kernel(float* __restrict__ deg, int n) {
    int i = blockIdx.x * blockDim.x + threadIdx.x;
    if (i < n) deg[i] = 1.0f;  // self loop
}

__global__ __launch_bounds__(256) void count_deg_kernel(const int* __restrict__ ei, int E,
                                                        float* __restrict__ deg) {
    int e = blockIdx.x * blockDim.x + threadIdx.x;
    if (e < E) atomicAdd(&deg[ei[E + e]], 1.0f);  // dst row of edge_index
}

__global__ __launch_bounds__(256) void dinv_kernel(const float* __restrict__ deg,
                                                   float* __restrict__ dinv, int n) {
    int i = blockIdx.x * blockDim.x + threadIdx.x;
    if (i < n) dinv[i] = rsqrtf(deg[i]);  // deg >= 1 guaranteed by self loop
}

// ---------------- WMMA GEMM: H = X @ W  (X: n x 128, W: 128 x 128) ----------------
// One wave computes one 16x16 output tile; 8 waves/block cover the 8 N-tiles of
// one M-tile (shared A rows hit in L0). K-loop: 32 x V_WMMA_F32_16X16X4_F32.

__global__ __launch_bounds__(256) void gemm_wmma_kernel(const float* __restrict__ X,
                                                        const float* __restrict__ W,
                                                        float* __restrict__ H, int n_rows) {
    const int lane = threadIdx.x & 31;
    const int wave = threadIdx.x >> 5;
    const int m0 = blockIdx.x * 16;        // one M-tile per block
    const int n0 = wave * 16;              // 8 N-tiles, one per wave
    if (m0 >= n_rows) return;

    const int half = lane >> 4;            // 0: lanes 0-15, 1: lanes 16-31
    const int l15  = lane & 15;

    // A frag: row = m0 + l15, K pair = k + 2*half (+0,+1)
    const float* arow = X + (size_t)(m0 + l15) * FEAT + 2 * half;
    // B frag: col = n0 + l15, K row = k + 2*half (+0,+1)
    const float* bcol = W + (size_t)(2 * half) * FEAT + (n0 + l15);

    v8f c = {};
#pragma unroll 4
    for (int k = 0; k < FEAT; k += 4) {
        v2f a;
        a.x = arow[k];
        a.y = arow[k + 1];
        v2f b;
        b.x = bcol[(size_t)k * FEAT];
        b.y = bcol[(size_t)(k + 1) * FEAT];
        // (neg_a, A, neg_b, B, c_mod, C, reuse_a, reuse_b)
        c = __builtin_amdgcn_wmma_f32_16x16x4_f32(false, a, false, b, (short)0, c,
                                                  false, false);
    }

    // D layout: VGPR r -> row m0 + r + 8*half, col n0 + l15
    float* out = H + (size_t)(m0 + 8 * half) * FEAT + n0 + l15;
#pragma unroll
    for (int r = 0; r < 8; ++r) out[(size_t)r * FEAT] = c[r];
}

// ---------------- aggregation ----------------

// acc[i][f] = h[i][f] * dinv[i]^2   (self-loop term; also deterministic init)
__global__ __launch_bounds__(256) void self_init_kernel(const float* __restrict__ h,
                                                        const float* __restrict__ dinv,
                                                        float* __restrict__ acc, int n) {
    size_t t = (size_t)blockIdx.x * blockDim.x + threadIdx.x;
    if (t >= (size_t)n * FEAT) return;
    int i = (int)(t >> 7);
    float di = dinv[i];
    acc[t] = h[t] * di * di;
}

// one wave per edge; each lane handles 4 consecutive features
__global__ __launch_bounds__(256) void edge_scatter_kernel(const int* __restrict__ ei, int E,
                                                           const float* __restrict__ dinv,
                                                           const float* __restrict__ h,
                                                           float* __restrict__ acc) {
    size_t tid = (size_t)blockIdx.x * blockDim.x + threadIdx.x;
    int e = (int)(tid >> 5);
    int lane = threadIdx.x & 31;
    if (e >= E) return;
    int s = ei[e];
    int d = ei[E + e];
    float w = dinv[s] * dinv[d];
    const float4* hs = (const float4*)(h + (size_t)s * FEAT);
    float4 v = hs[lane];
    float* out = acc + (size_t)d * FEAT + lane * 4;
    atomicAdd(out + 0, v.x * w);
    atomicAdd(out + 1, v.y * w);
    atomicAdd(out + 2, v.z * w);
    atomicAdd(out + 3, v.w * w);
}

// x_out[i][f] = relu(acc[i][f] + b[f])
__global__ __launch_bounds__(256) void finish_kernel(const float* __restrict__ acc,
                                                     const float* __restrict__ b,
                                                     float* __restrict__ xout, int n) {
    size_t t = (size_t)blockIdx.x * blockDim.x + threadIdx.x;
    if (t >= (size_t)n * FEAT) return;
    float v = acc[t] + b[t & (FEAT - 1)];
    xout[t] = v > 0.0f ? v : 0.0f;
}

// ---------------- launch ----------------

extern "C" void kernel_launch(void* const* d_in, const int* in_sizes, int n_in,
                              void* d_out, int out_size, void* d_ws, size_t ws_size,
                              hipStream_t stream) {
    const float* x  = (const float*)d_in[0];
    const int*   ei = (const int*)d_in[1];
    const float* Wl[3] = {(const float*)d_in[2], (const float*)d_in[4], (const float*)d_in[6]};
    const float* bl[3] = {(const float*)d_in[3], (const float*)d_in[5], (const float*)d_in[7]};
    float* out = (float*)d_out;

    const int N = in_sizes[0] / FEAT;  // 100000
    const int E = in_sizes[1] / 2;     // 3200000

    float* ws   = (float*)d_ws;
    float* deg  = ws;                          // N
    float* dinv = ws + N;                      // N
    float* h    = ws + 2 * (size_t)N;          // N*128
    float* acc  = h + (size_t)N * FEAT;        // N*128
    float* xb   = acc + (size_t)N * FEAT;      // N*128

    // degrees + symmetric normalization (edges fixed across layers)
    init_deg_kernel<<<(N + 255) / 256, 256, 0, stream>>>(deg, N);
    count_deg_kernel<<<(E + 255) / 256, 256, 0, stream>>>(ei, E, deg);
    dinv_kernel<<<(N + 255) / 256, 256, 0, stream>>>(deg, dinv, N);

    const int elem_blocks = (int)(((size_t)N * FEAT + 255) / 256);
    const int edge_blocks = (int)(((size_t)E * 32 + 255) / 256);
    const int gemm_blocks = (N + 15) / 16;  // one M-tile per block (8 waves)

    const float* cur = x;
    for (int l = 0; l < 3; ++l) {
        gemm_wmma_kernel<<<gemm_blocks, 256, 0, stream>>>(cur, Wl[l], h, N);
        self_init_kernel<<<elem_blocks, 256, 0, stream>>>(h, dinv, acc, N);
        edge_scatter_kernel<<<edge_blocks, 256, 0, stream>>>(ei, E, dinv, h, acc);
        float* dst = (l == 2) ? out : xb;
        finish_kernel<<<elem_blocks, 256, 0, stream>>>(acc, bl[l], dst, N);
        cur = xb;
    }
}